// ChamferDistKDTree_47906065219739
// MI455X (gfx1250) — compile-verified
//
#include <hip/hip_runtime.h>
#include <hip/hip_bf16.h>
#include <math.h>

// ---------------------------------------------------------------------------
// Chamfer distance, B=8, M=N=4096, 3-D points.
// Metric trick: d2(q,r) = |q|^2 + |r|^2 - 2 q.r.  Per-query argmin over r only
// needs  m(q,r) = |r|^2 - 2 q.r, which is an exact 16x16x4 f32 WMMA with
//   A row = (qx, qy, qz, 1)        (16 queries  x K=4)
//   B col = (-2rx, -2ry, -2rz, |r|^2)  (K=4 x 16 refs)
// Final distance = sqrt(max(0, min_m + |q|^2)).  No argmin index needed.
// ---------------------------------------------------------------------------

typedef __attribute__((ext_vector_type(2))) float v2f;
typedef __attribute__((ext_vector_type(8))) float v8f;

#define NBATCH 8
#define NPTS   4096
#define WAVES  4                 // waves per block (wave32)
#define BLOCK  (WAVES * 32)      // 128 threads
#define QPB    (WAVES * 16)      // 64 queries per block (16 per wave)
#define CHUNK  2048              // refs staged per LDS chunk (32 KB)

__global__ __launch_bounds__(BLOCK)
void chamfer_nn_kernel(const float* __restrict__ tar,
                       const float* __restrict__ src,
                       float* __restrict__ out)   // [2][NBATCH][NPTS]: acc, comp
{
    const int b   = blockIdx.y;
    const int dir = blockIdx.z;          // 0: query=src, ref=tar (accuracy)
                                         // 1: query=tar, ref=src (complete)
    const float* qry = (dir == 0 ? src : tar) + b * NPTS * 3;
    const float* ref = (dir == 0 ? tar : src) + b * NPTS * 3;
    float* o = out + dir * (NBATCH * NPTS) + b * NPTS;

    const int tid  = threadIdx.x;
    const int wave = tid >> 5;
    const int lane = tid & 31;
    const int hi   = lane >> 4;          // half of wave: selects K-pair {0,1} vs {2,3}
    const int col  = lane & 15;          // tile column / query-row loader index

    // Augmented refs in LDS as float2 pairs: [2*i+0]=(-2x,-2y) [2*i+1]=(-2z,|r|^2)
    __shared__ float2 s_ref[2 * CHUNK];

    // ---- A operand: 16 queries for this wave, fixed for the whole kernel ----
    const int qbase = blockIdx.x * QPB + wave * 16;
    const float* qp = qry + (qbase + col) * 3;
    const float qx = qp[0], qy = qp[1], qz = qp[2];
    v2f Av;
    Av.x = hi ? qz : qx;                 // K=2 : K=0
    Av.y = hi ? 1.0f : qy;               // K=3 : K=1

    // Running min of metric tile rows (8 accumulator VGPRs of D layout).
    float bm[8];
#pragma unroll
    for (int r = 0; r < 8; ++r) bm[r] = 3.4e38f;

    const v8f czero = {};

    for (int c0 = 0; c0 < NPTS; c0 += CHUNK) {
        __syncthreads();
        // Cooperative augmented stage: 2048 refs / 128 threads = 16 each.
        for (int i = tid; i < CHUNK; i += BLOCK) {
            const float* rp = ref + (c0 + i) * 3;
            const float x = rp[0], y = rp[1], z = rp[2];
            s_ref[2 * i + 0] = make_float2(-2.0f * x, -2.0f * y);
            s_ref[2 * i + 1] = make_float2(-2.0f * z, x * x + y * y + z * z);
        }
        __syncthreads();

#pragma unroll 4
        for (int t = 0; t < CHUNK; t += 16) {
            // B operand: one conflict-free ds_load_b64 per lane.
            const float2 bv = s_ref[2 * (t + col) + hi];
            v2f Bv;
            Bv.x = bv.x;
            Bv.y = bv.y;
            // D[16x16] = |r|^2 - 2 q.r for this 16x16 (query x ref) tile.
            v8f d = __builtin_amdgcn_wmma_f32_16x16x4_f32(
                false, Av, false, Bv, (short)0, czero, false, false);
#pragma unroll
            for (int r = 0; r < 8; ++r) bm[r] = fminf(bm[r], d[r]);
        }
    }

    // ---- Cross-lane min within each 16-lane half (D: lanes0-15 -> M=r,
    //      lanes16-31 -> M=r+8).  xor masks 1..8 stay inside the half. ----
#pragma unroll
    for (int r = 0; r < 8; ++r) {
        float v = bm[r];
        v = fminf(v, __shfl_xor(v, 1, 32));
        v = fminf(v, __shfl_xor(v, 2, 32));
        v = fminf(v, __shfl_xor(v, 4, 32));
        v = fminf(v, __shfl_xor(v, 8, 32));
        bm[r] = v;
    }

    if (col == 0) {                      // lane 0 (queries qbase+0..7) and
        const int qb = qbase + hi * 8;   // lane 16 (queries qbase+8..15)
#pragma unroll
        for (int r = 0; r < 8; ++r) {
            const float* q2p = qry + (qb + r) * 3;
            const float x = q2p[0], y = q2p[1], z = q2p[2];
            const float d2 = bm[r] + (x * x + y * y + z * z);
            o[qb + r] = sqrtf(fmaxf(d2, 0.0f));
        }
    }
}

__global__ __launch_bounds__(256)
void chamfer_reduce_kernel(const float* __restrict__ acc,
                           const float* __restrict__ comp,
                           float* __restrict__ cham)
{
    const int b = blockIdx.x;
    __shared__ float red[256];
    float s = 0.0f;
    for (int i = threadIdx.x; i < NPTS; i += 256)
        s += acc[b * NPTS + i] + comp[b * NPTS + i];
    red[threadIdx.x] = s;
    __syncthreads();
    for (int st = 128; st > 0; st >>= 1) {
        if (threadIdx.x < st) red[threadIdx.x] += red[threadIdx.x + st];
        __syncthreads();
    }
    if (threadIdx.x == 0)
        cham[b] = red[0] / (2.0f * (float)NPTS);   // 0.5*(meanA+meanC)
}

extern "C" void kernel_launch(void* const* d_in, const int* in_sizes, int n_in,
                              void* d_out, int out_size, void* d_ws, size_t ws_size,
                              hipStream_t stream) {
    (void)in_sizes; (void)n_in; (void)d_ws; (void)ws_size; (void)out_size;
    const float* tar = (const float*)d_in[0];   // [8,4096,3]
    const float* src = (const float*)d_in[1];   // [8,4096,3]
    float* out = (float*)d_out;                 // acc[8,4096] | comp[8,4096] | cham[8]

    dim3 grid(NPTS / QPB, NBATCH, 2);           // 64 x 8 x 2 blocks
    chamfer_nn_kernel<<<grid, BLOCK, 0, stream>>>(tar, src, out);

    const float* acc  = out;
    const float* comp = out + NBATCH * NPTS;
    float* cham       = out + 2 * NBATCH * NPTS;
    chamfer_reduce_kernel<<<NBATCH, 256, 0, stream>>>(acc, comp, cham);
}